// CellPathwayPoolingAggregator_72782515798453
// MI455X (gfx1250) — compile-verified
//
#include <hip/hip_runtime.h>
#include <stdint.h>

#define TPB    256
#define CELLS  4        // cells (rows of x) staged per block
#define MAX_G  10240    // genes; 4 rows interleaved = 160 KB LDS (ref G = 10000)

// ---------------------------------------------------------------------------
// Main kernel: one block per group of 4 cells. Stage the 4 rows of x into LDS
// gene-interleaved via CDNA5 async global->LDS loads (ASYNCcnt path), then
// each thread reduces pathways: one ds_load_b128 per gene index yields the
// 4 cells' values at once; flat-index loads are amortized 4x.
// ---------------------------------------------------------------------------
__global__ __launch_bounds__(TPB)
void pool_lds_kernel(const float* __restrict__ x,
                     const int*   __restrict__ flat,
                     const int*   __restrict__ offs,
                     const float* __restrict__ sizes,
                     float* __restrict__ out,
                     int B, int G, int P) {
    __shared__ float4 smem[MAX_G];          // smem[g] = {x[b0+0,g], .., x[b0+3,g]}

    const int b0  = blockIdx.x * CELLS;
    const int tid = threadIdx.x;
    const int nc  = (B - b0 < CELLS) ? (B - b0) : CELLS;

    // Low 32 bits of a flat shared-memory address are the wave-relative LDS
    // byte offset (ISA section 10.2 aperture rules).
    const uint32_t lds0 = (uint32_t)(uintptr_t)(void*)&smem[0];

    // --- async stage: per (gene, cell) dword; lanes sweep consecutive genes
    //     of one row, so global reads are fully coalesced (128 B per wave).
    //     LDS destination is interleaved: byte offset g*16 + i*4.
    for (int i = 0; i < nc; ++i) {
        const uint64_t rb = (uint64_t)(uintptr_t)(x + (size_t)(b0 + i) * (size_t)G);
        const uint32_t li = lds0 + (uint32_t)i * 4u;
        for (uint32_t g = tid; g < (uint32_t)G; g += TPB) {
            uint32_t l  = li + g * 16u;
            uint64_t ga = rb + (uint64_t)g * 4u;
            asm volatile("global_load_async_to_lds_b32 %0, %1, off"
                         :: "v"(l), "v"(ga) : "memory");
        }
    }
    asm volatile("s_wait_asynccnt 0" ::: "memory");   // this wave's asyncs done
    __syncthreads();                                  // all waves' LDS writes visible

    // --- segmented mean: each thread owns pathways p = tid, tid+TPB, ... ---
    for (int p = tid; p < P; p += TPB) {
        const int   beg = offs[p];
        const float sz  = sizes[p];
        const int   end = beg + (int)(sz + 0.5f);
        float a0 = 0.f, a1 = 0.f, a2 = 0.f, a3 = 0.f;
        #pragma unroll 4
        for (int k = beg; k < end; ++k) {
            const float4 v = smem[flat[k]];           // one ds_load_b128 -> 4 cells
            a0 += v.x; a1 += v.y; a2 += v.z; a3 += v.w;
        }
        const float inv = 1.0f / sz;
        float acc[CELLS] = {a0, a1, a2, a3};
        for (int i = 0; i < nc; ++i)
            out[(size_t)(b0 + i) * (size_t)P + p] = acc[i] * inv;
    }
}

// ---------------------------------------------------------------------------
// Exclusive prefix-sum of segment sizes -> per-pathway start offsets.
// segment_ids from the reference are sorted (np.repeat(arange)), so pathway p
// owns flat_indices[offs[p] .. offs[p]+size_p).
// ---------------------------------------------------------------------------
__global__ __launch_bounds__(1024)
void seg_offsets_kernel(const float* __restrict__ sizes, int* __restrict__ offs, int P) {
    __shared__ int s[1024];
    const int tid = threadIdx.x;

    if (P > 1024) {               // generic fallback (not hit for P=1000)
        if (tid == 0) {
            int a = 0;
            for (int p = 0; p < P; ++p) { offs[p] = a; a += (int)(sizes[p] + 0.5f); }
        }
        return;
    }

    const int v = (tid < P) ? (int)(sizes[tid] + 0.5f) : 0;
    s[tid] = v;
    __syncthreads();
    for (int d = 1; d < 1024; d <<= 1) {        // Hillis-Steele inclusive scan
        int t = (tid >= d) ? s[tid - d] : 0;
        __syncthreads();
        s[tid] += t;
        __syncthreads();
    }
    if (tid < P) offs[tid] = s[tid] - v;        // exclusive
}

// ---------------------------------------------------------------------------
// Fallback (G too large for the LDS buffer): direct global gather, one thread
// per (b, p). Not taken for the reference shapes.
// ---------------------------------------------------------------------------
__global__ __launch_bounds__(TPB)
void pool_direct_kernel(const float* __restrict__ x,
                        const int*   __restrict__ flat,
                        const int*   __restrict__ offs,
                        const float* __restrict__ sizes,
                        float* __restrict__ out,
                        long long total, int G, int P) {
    long long idx = (long long)blockIdx.x * TPB + threadIdx.x;
    if (idx >= total) return;
    const int p = (int)(idx % P);
    const long long b = idx / P;
    const float* xr = x + (size_t)b * (size_t)G;
    const int   beg = offs[p];
    const float sz  = sizes[p];
    const int   end = beg + (int)(sz + 0.5f);
    float acc = 0.0f;
    for (int k = beg; k < end; ++k) acc += xr[flat[k]];
    out[idx] = acc / sz;
}

extern "C" void kernel_launch(void* const* d_in, const int* in_sizes, int n_in,
                              void* d_out, int out_size, void* d_ws, size_t ws_size,
                              hipStream_t stream) {
    const float* x     = (const float*)d_in[0];   // (B, G) f32
    const int*   flat  = (const int*)  d_in[1];   // (N,)   i32
    // d_in[2] = segment_ids (sorted; offsets recomputed from sizes instead)
    const float* sizes = (const float*)d_in[3];   // (P,)   f32
    float* out = (float*)d_out;

    const int P = in_sizes[3];
    const int B = out_size / P;
    const int G = in_sizes[0] / B;

    int* offs = (int*)d_ws;                       // P ints of scratch

    seg_offsets_kernel<<<1, 1024, 0, stream>>>(sizes, offs, P);

    if (G <= MAX_G) {
        const int blocks = (B + CELLS - 1) / CELLS;
        pool_lds_kernel<<<blocks, TPB, 0, stream>>>(x, flat, offs, sizes, out, B, G, P);
    } else {
        const long long total = (long long)B * P;
        const int nblk = (int)((total + TPB - 1) / TPB);
        pool_direct_kernel<<<nblk, TPB, 0, stream>>>(x, flat, offs, sizes, out,
                                                     total, G, P);
    }
}